// GraphLearningModule_53626961658417
// MI455X (gfx1250) — compile-verified
//
#include <hip/hip_runtime.h>
#include <cstdint>
#include <cstddef>

// Problem constants (from reference)
#define NBATCH 32
#define NNODE  50000
#define KNBR   32
#define EDIM   6
#define FDIM   3
#define THETA_ 0.5f
#define NEG_SLOPE_ 0.2f

typedef __attribute__((ext_vector_type(2))) float v2f;
typedef __attribute__((ext_vector_type(8))) float v8f;

// ---------------------------------------------------------------------------
// Phase 1: f[row=n*32+b][f] = leaky_relu((x[b,n]+emb[n,:]) . W[f,:] + bias[f])
// Real 16x16x4 f32 WMMA: each wave produces 16 rows.
//   A (16x4 f32): lanes 0-15 -> M=lane,    VGPR0=K0, VGPR1=K1
//                 lanes 16-31 -> M=lane-16, VGPR0=K2, VGPR1=K3
//   B (4x16 f32): same striping with N in place of M; B[k][ncol] = W[ncol][k]
//   D (16x16 f32): VGPR j: lanes 0-15 -> M=j,   N=lane
//                          lanes 16-31 -> M=j+8, N=lane-16
// Two chained WMMAs cover K = 0..3 and 4..5 (6,7 zero-padded).
// Rows are stored padded to 4 floats (16B) so phase 2 loads are b128-aligned.
// ---------------------------------------------------------------------------
__global__ __launch_bounds__(256)
void gl_phase1_wmma(const float* __restrict__ x,      // [B][N]
                    const float* __restrict__ emb,    // [N][E]
                    const float* __restrict__ W,      // [F][E]
                    const float* __restrict__ bias,   // [F]
                    float* __restrict__ fbuf)         // [N*B][4]
{
    const int  lane = threadIdx.x & 31;
    const long wave = (long)blockIdx.x * (blockDim.x >> 5) + (threadIdx.x >> 5);
    const long rowBase = wave * 16;
    const long NB = (long)NNODE * NBATCH;
    if (rowBase >= NB) return;               // wave-uniform: EXEC stays all-ones

    const int  M   = lane & 15;
    const bool hi  = lane >= 16;
    const long row = rowBase + M;            // row = n*32 + b
    const int  n   = (int)(row >> 5);
    const int  b   = (int)(row & 31);

    const float xv = x[(long)b * NNODE + n];

    // A chunk 1: K = {0,1} (lanes<16) / {2,3} (lanes>=16)
    const int kb1 = hi ? 2 : 0;
    v2f a1;
    a1.x = xv + emb[n * EDIM + kb1 + 0];
    a1.y = xv + emb[n * EDIM + kb1 + 1];
    // A chunk 2: K = {4,5} (lanes<16) / {6,7}=0 (lanes>=16)
    v2f a2 = {0.0f, 0.0f};
    if (!hi) {
        a2.x = xv + emb[n * EDIM + 4];
        a2.y = xv + emb[n * EDIM + 5];
    }

    // B = W^T, only first FDIM of 16 columns populated
    const int Nc = lane & 15;
    v2f b1 = {0.0f, 0.0f}, b2 = {0.0f, 0.0f};
    if (Nc < FDIM) {
        b1.x = W[Nc * EDIM + kb1 + 0];
        b1.y = W[Nc * EDIM + kb1 + 1];
        if (!hi) {
            b2.x = W[Nc * EDIM + 4];
            b2.y = W[Nc * EDIM + 5];
        }
    }

    v8f c = {};
    c = __builtin_amdgcn_wmma_f32_16x16x4_f32(false, a1, false, b1,
                                              (short)0, c, false, false);
    c = __builtin_amdgcn_wmma_f32_16x16x4_f32(false, a2, false, b2,
                                              (short)0, c, false, false);

    // bias + leaky_relu + scatter rows (only 6 lanes carry useful columns)
    if (Nc < FDIM) {
        const float bi = bias[Nc];
        const long  rb = rowBase + (hi ? 8 : 0);
#pragma unroll
        for (int j = 0; j < 8; ++j) {
            float v = c[j] + bi;
            v = v > 0.0f ? v : NEG_SLOPE_ * v;
            fbuf[(rb + j) * 4 + Nc] = v;
        }
    }
}

// ---------------------------------------------------------------------------
// Phase 2: one wave per node n; lane = batch b (wave32 == B exactly).
//   Self row:      coalesced b128 load (512B/wave).
//   Neighbor rows: readlane-broadcast index -> coalesced b128 gather from L2
//                  (fbuf = 25.6MB, resident in the 192MB L2).
//   mean over b:   multi-value butterfly (transpose-reduce): 31 shuffles total
//                  reduce 32 values x 32 lanes so lane k ends with sum for k.
// ---------------------------------------------------------------------------
__global__ __launch_bounds__(256)
void gl_phase2(const float4* __restrict__ fbuf,   // [N*32]
               const int*    __restrict__ nbr,    // [N][K]
               float*        __restrict__ out)    // [N][K]
{
    const int  lane = threadIdx.x & 31;
    const long wave = (long)blockIdx.x * (blockDim.x >> 5) + (threadIdx.x >> 5);
    if (wave >= NNODE) return;                    // wave-uniform
    const int node = (int)wave;

    const int my_nbr = nbr[(long)node * KNBR + lane];
    const float4 fs = fbuf[(long)node * 32 + lane];

    const float inv2theta = 1.0f / (2.0f * THETA_);

    // vals[k] = exp(-||f_self - f_nbr(k)||^2 / (2*theta)), one per neighbor,
    // each distributed over the 32 lanes (= batch b).
    float vals[KNBR];
#pragma unroll
    for (int k = 0; k < KNBR; ++k) {
        int m = __builtin_amdgcn_readlane(my_nbr, k);
        m = m < 0 ? 0 : m;                        // guard OOB (masked later)
        const float4 fn = fbuf[(long)m * 32 + lane];
        const float d0 = fs.x - fn.x;
        const float d1 = fs.y - fn.y;
        const float d2 = fs.z - fn.z;
        const float d2s = d0 * d0 + d1 * d1 + d2 * d2;
        vals[k] = __expf(-d2s * inv2theta);
    }

    // Multi-value butterfly reduce: level o pairs (vals[i], vals[i+o]) across
    // lane-offset o. After all levels, lane L holds sum over b of vals[L].
    // Total: 16+8+4+2+1 = 31 shuffles (vs 160 for per-k butterflies).
#pragma unroll
    for (int o = 16; o >= 1; o >>= 1) {
#pragma unroll
        for (int i = 0; i < 16; ++i) {
            if (i < o) {
                const float u = vals[i];
                const float w = vals[i + o];
                const float keep = (lane & o) ? w : u;
                const float send = (lane & o) ? u : w;
                vals[i] = keep + __shfl_xor(send, o, 32);
            }
        }
    }

    float res = vals[0] * (1.0f / 32.0f);
    if (my_nbr < 0) res = 0.0f;                   // mask (nbr >= 0)
    out[(long)node * KNBR + lane] = res;
}

// ---------------------------------------------------------------------------
extern "C" void kernel_launch(void* const* d_in, const int* in_sizes, int n_in,
                              void* d_out, int out_size, void* d_ws, size_t ws_size,
                              hipStream_t stream) {
    const float* x    = (const float*)d_in[0];   // (32, 50000)
    const float* emb  = (const float*)d_in[1];   // (50000, 6)
    const float* W    = (const float*)d_in[2];   // (3, 6)
    const float* bias = (const float*)d_in[3];   // (3,)
    const int*   nbr  = (const int*)  d_in[4];   // (50000, 32)
    float*       out  = (float*)d_out;           // (50000, 32)
    float*       fbuf = (float*)d_ws;            // N*B*4 floats = 25.6 MB

    // Phase 1: N*B/16 = 100000 waves, 8 waves per 256-thread block
    const long NB = (long)NNODE * NBATCH;
    const int  p1_waves  = (int)(NB / 16);
    const int  p1_blocks = (p1_waves + 7) / 8;
    gl_phase1_wmma<<<p1_blocks, 256, 0, stream>>>(x, emb, W, bias, fbuf);

    // Phase 2: one wave per node
    const int p2_blocks = (NNODE + 7) / 8;
    gl_phase2<<<p2_blocks, 256, 0, stream>>>((const float4*)fbuf, nbr, out);
}